// LoadingModel_39565238730854
// MI455X (gfx1250) — compile-verified
//
#include <hip/hip_runtime.h>

#define NDIM 1000
#define PDIM 50000
#define ZDIM 10
#define LDIM 5
#define NSTEP (ZDIM * LDIM)          // 50 sequential (k,l) updates
#define ZP    (ZDIM * PDIM)          // 500000
#define NEG_FLT_MAX_F (-3.402823466e+38f)

// ZtD GEMM tiling
#define COLS_PER_BLOCK 128           // 8 waves x 16-wide WMMA tiles
#define KCHUNK 40                    // K rows per TDM tile (1000 = 25 * 40)
#define NCHUNK (NDIM / KCHUNK)       // 25
#define ROWDW  136                   // 128 data dwords + 8 pad dwords (TDM pad) -> bank-conflict-free

typedef float v2f __attribute__((ext_vector_type(2)));
typedef float v8f __attribute__((ext_vector_type(8)));
typedef unsigned int v4u __attribute__((ext_vector_type(4)));
typedef int v4i __attribute__((ext_vector_type(4)));
typedef int v8i __attribute__((ext_vector_type(8)));

// ---------------------------------------------------------------------------
// D# descriptor packing (CDNA5 ISA ch.8). Computed OUTSIDE the wave-0 guard so
// uniformity analysis keeps it in SALU/SGPRs (avoids v_readfirstlane storms).
// group0 = {count/type, lds_addr, global_addr}; group1 = dims/strides + LDS
// padding (8 dwords after every 128 -> row stride 136 dwords, so the two WMMA
// half-waves hit disjoint LDS bank groups). tensor_dim = remaining extent from
// tile start so the ragged last column block reads architected zeros.
// ---------------------------------------------------------------------------
__device__ __forceinline__ v4u tdm_g0(unsigned lds_addr, const float* gaddr) {
  const unsigned long long ga = (unsigned long long)(uintptr_t)gaddr;
  v4u g0;
  g0.x = 1u;                                                // count=1 (valid), user mode
  g0.y = lds_addr;                                          // LDS byte address
  g0.z = (unsigned)(ga & 0xffffffffu);                      // global_addr lo
  g0.w = (unsigned)((ga >> 32) & 0x01ffffffu) | (2u << 30); // global_addr hi | type=2
  return g0;
}

__device__ __forceinline__ v8i tdm_g1(unsigned tensor_d0, unsigned tensor_d1) {
  v8i g1;
  g1[0] = (int)((2u << 16)        // data_size = 4B
              | (1u << 20)        // pad_enable
              | (6u << 22)        // pad_interval: 128 dwords
              | (7u << 25));      // pad_amount: 8 dwords
  g1[1] = (int)((tensor_d0 & 0xffffu) << 16);                       // tensor_dim0 lo16
  g1[2] = (int)((tensor_d0 >> 16) | ((tensor_d1 & 0xffffu) << 16)); // dim0 hi | dim1 lo
  g1[3] = (int)((tensor_d1 >> 16) | ((unsigned)COLS_PER_BLOCK << 16)); // dim1 hi | tile_dim0
  g1[4] = KCHUNK;                 // tile_dim1 | tile_dim2=0 (2D)
  g1[5] = PDIM;                   // tensor_dim0_stride lo
  g1[6] = 0;                      // stride0 hi | tensor_dim1_stride lo (unused, 2D)
  g1[7] = 0;
  return g1;
}

__device__ __forceinline__ void tdm_issue(v4u g0, v8i g1) {
  const v4i gz = {0, 0, 0, 0};
#if __clang_major__ >= 23
  const v8i gz8 = {0, 0, 0, 0, 0, 0, 0, 0};
  __builtin_amdgcn_tensor_load_to_lds(g0, g1, gz, gz, gz8, 0);
#else
  __builtin_amdgcn_tensor_load_to_lds(g0, g1, gz, gz, 0);
#endif
}

// ---------------------------------------------------------------------------
// Phase A: ZtD = mean_z^T @ data  [Z=10 (pad 16), P=50000, K=1000].
// B tiles arrive via double-buffered TDM; A fragments are one b64 load from a
// pre-padded pair table (L2-resident). 250 V_WMMA_F32_16X16X4_F32 per wave.
// Memory bound: 200MB streamed once (~8.6us at 23.3 TB/s).
// ---------------------------------------------------------------------------
__global__ void __launch_bounds__(256)
ztd_wmma_tdm_kernel(const float* __restrict__ data,
                    const float* __restrict__ mzPair,   // [500][16][2] zero-padded
                    float* __restrict__ ztd) {
  __shared__ float bbuf[2][KCHUNK * ROWDW];

  const int lane    = threadIdx.x & 31;
  const int wave    = threadIdx.x >> 5;
  const int colbase = blockIdx.x * COLS_PER_BLOCK;
  const int m       = lane & 15;
  const int khalf   = (lane >> 4) << 1;    // 0 or 2
  const int wcol    = wave << 4;
  const unsigned td0 = (unsigned)(PDIM - colbase);   // remaining columns

  // Prologue: descriptor built uniformly, issued by wave 0 only.
  {
    const v4u g0 = tdm_g0((unsigned)(uintptr_t)&bbuf[0][0], data + colbase);
    const v8i g1 = tdm_g1(td0, (unsigned)NDIM);
    if (threadIdx.x < 32) tdm_issue(g0, g1);
  }

  v8f acc = {};
  for (int c = 0; c < NCHUNK; ++c) {
    if (c + 1 < NCHUNK) {                  // prefetch next tile, then wait on older
      const v4u g0 = tdm_g0((unsigned)(uintptr_t)&bbuf[(c + 1) & 1][0],
                            data + (size_t)(c + 1) * KCHUNK * PDIM + colbase);
      const v8i g1 = tdm_g1(td0, (unsigned)(NDIM - (c + 1) * KCHUNK));
      if (threadIdx.x < 32) {
        tdm_issue(g0, g1);
        __builtin_amdgcn_s_wait_tensorcnt((short)1);
      }
    } else {
      if (threadIdx.x < 32) __builtin_amdgcn_s_wait_tensorcnt((short)0);
    }
    __syncthreads();                       // release tile to all 8 waves
    const float* buf = &bbuf[c & 1][0];
#pragma unroll
    for (int kk = 0; kk < KCHUNK; kk += 4) {
      const int k0 = kk + khalf;
      v2f b, a;
      // B[k, n] from LDS; half-waves hit disjoint bank groups (stride 136 dw)
      b.x = buf[k0 * ROWDW + wcol + m];
      b.y = buf[(k0 + 1) * ROWDW + wcol + m];
      // A[m=z, k=n] pair: single unpredicated 8B load, zero-padded for z>=10
      const int q = c * (KCHUNK / 2) + (kk >> 1) + (lane >> 4);
      a = *(const v2f*)(mzPair + ((size_t)q * 16 + m) * 2);
      acc = __builtin_amdgcn_wmma_f32_16x16x4_f32(false, a, false, b,
                                                  (short)0, acc, false, false);
    }
    __syncthreads();                       // all reads done before TDM reuses buffer
  }

  const int pcol = colbase + wcol + m;
  const int zrow = (lane < 16) ? 0 : 8;
#pragma unroll
  for (int r = 0; r < 8; ++r) {
    const int z = zrow + r;
    if (z < ZDIM && pcol < PDIM) ztd[(size_t)z * PDIM + pcol] = acc[r];
  }
}

// ---------------------------------------------------------------------------
// mean_zz = mean_z^T @ mean_z + N * var_z  (10x10 in one 16x16 WMMA tile).
// Symmetric Gram: A-fragment == B-fragment, from the padded pair table.
// ---------------------------------------------------------------------------
__global__ void mzz_wmma_kernel(const float* __restrict__ mzPair,
                                const float* __restrict__ varz,
                                float* __restrict__ mzz) {
  const int lane = threadIdx.x & 31;
  const int m    = lane & 15;
  v8f acc = {};
  for (int kb = 0; kb < NDIM; kb += 4) {
    const int q = (kb >> 1) + (lane >> 4);
    const v2f a = *(const v2f*)(mzPair + ((size_t)q * 16 + m) * 2);
    acc = __builtin_amdgcn_wmma_f32_16x16x4_f32(false, a, false, a,
                                                (short)0, acc, false, false);
  }
  const int zrow = (lane < 16) ? 0 : 8;
#pragma unroll
  for (int r = 0; r < 8; ++r) {
    const int z = zrow + r;
    if (z < ZDIM && m < ZDIM)
      mzz[z * ZDIM + m] = acc[r] + (float)NDIM * varz[z * ZDIM + m];
  }
}

// ---------------------------------------------------------------------------
// Init: W0[z,p] = sum_l mean_w*alpha ; zero-padded A pair table ; reset the
// per-step (max,sumexp) slots. Runs every call -> graph-replay safe.
// ---------------------------------------------------------------------------
__global__ void init_kernel(const float* __restrict__ mz,
                            const float* __restrict__ mw_in,
                            const float* __restrict__ al_in,
                            float* __restrict__ W,
                            float* __restrict__ mzPair,
                            unsigned long long* __restrict__ msSlots) {
  const int idx = blockIdx.x * blockDim.x + threadIdx.x;
  if (idx < NSTEP) {
    // low 32b = running max (-FLT_MAX), high 32b = sumexp (0.0f bits == 0)
    msSlots[idx] = (unsigned long long)__float_as_uint(NEG_FLT_MAX_F);
  }
  if (idx < (NDIM / 2) * 16) {             // 8000 pair entries
    const int q = idx >> 4, mm = idx & 15;
    float x0 = 0.0f, x1 = 0.0f;
    if (mm < ZDIM) {
      x0 = mz[(size_t)(2 * q) * ZDIM + mm];
      x1 = mz[(size_t)(2 * q + 1) * ZDIM + mm];
    }
    mzPair[(size_t)idx * 2 + 0] = x0;
    mzPair[(size_t)idx * 2 + 1] = x1;
  }
  if (idx < ZP) {
    float acc = 0.0f;
#pragma unroll
    for (int l = 0; l < LDIM; ++l)
      acc += mw_in[(size_t)l * ZP + idx] * al_in[(size_t)l * ZP + idx];
    W[idx] = acc;
  }
}

__device__ __forceinline__ void softmax_merge(float& m, float& s, float om, float os) {
  const float nm = fmaxf(m, om);
  s = s * __expf(m - nm) + os * __expf(om - nm);   // exp(-huge)->0, no NaN paths
  m = nm;
}

// ---------------------------------------------------------------------------
// One fused sequential step. s in [0,50]:
//   (a) finalize step s-1: alpha/u_mean from merged (m,S); write mw/vw/al/W-row
//   (b) if new k: RtZk[p] = ZtD[k,p] - sum_{j!=k} mzz[k,j]*W[j,p]
//   (c) logits for step s + wave32 online-softmax + atomicCAS merge into slot s
// ---------------------------------------------------------------------------
__global__ void vb_step_kernel(int s,
                               const float* __restrict__ mw_in,
                               const float* __restrict__ al_in,
                               const float* __restrict__ tau0_in,
                               const float* __restrict__ pi_in,
                               const float* __restrict__ tau_in,
                               float* __restrict__ W,
                               const float* __restrict__ ZtD,
                               float* __restrict__ RtZk,
                               float* __restrict__ logits,
                               const float* __restrict__ mzz,
                               unsigned long long* __restrict__ msSlots,
                               float* __restrict__ mw_out,
                               float* __restrict__ vw_out,
                               float* __restrict__ al_out) {
  __shared__ float shm[8];
  __shared__ float shs[8];

  const int  p     = blockIdx.x * blockDim.x + threadIdx.x;
  const bool valid = (p < PDIM);
  const float tau  = tau_in[0];

  float WkReg = 0.0f;   // freshly updated W row (row kp)
  int   kp    = -1;

  // ---- (a) finalize previous step -------------------------------------
  if (s > 0) {
    const int sp = s - 1;
    kp = sp / LDIM;
    const int lp = sp % LDIM;
    const float Ezzp  = mzz[kp * ZDIM + kp];
    const float uvarp = 1.0f / (tau * Ezzp + tau0_in[lp * ZDIM + kp]);
    if (valid) {
      const unsigned long long msv = msSlots[sp];
      const float mM = __uint_as_float((unsigned)(msv & 0xffffffffu));
      const float mS = __uint_as_float((unsigned)(msv >> 32));
      const float logit = logits[p];
      const float alpha = __expf(logit - mM) / mS;
      const size_t off  = ((size_t)lp * ZDIM + kp) * PDIM + p;
      const float Wkl   = W[(size_t)kp * PDIM + p] - mw_in[off] * al_in[off];
      const float ERtZ  = RtZk[p] - Ezzp * Wkl;
      const float um    = tau * uvarp * ERtZ;
      mw_out[off] = um;
      al_out[off] = alpha;
      WkReg = Wkl + um * alpha;
      W[(size_t)kp * PDIM + p] = WkReg;
    }
    if (p == 0) vw_out[lp * ZDIM + kp] = uvarp;
  }

  // ---- (b)+(c) current step's logits + softmax statistics -------------
  float mLoc = NEG_FLT_MAX_F, sLoc = 0.0f;
  if (s < NSTEP) {
    const int k = s / LDIM, l = s % LDIM;
    const float Ezz = mzz[k * ZDIM + k];
    if (valid) {
      float rtz, Wk;
      if (l == 0) {
        rtz = ZtD[(size_t)k * PDIM + p];
#pragma unroll
        for (int j = 0; j < ZDIM; ++j) {
          if (j == k) continue;
          const float Wj = (j == kp) ? WkReg : W[(size_t)j * PDIM + p];
          rtz -= mzz[k * ZDIM + j] * Wj;
        }
        RtZk[p] = rtz;
        Wk = W[(size_t)k * PDIM + p];
      } else {
        rtz = RtZk[p];
        Wk  = WkReg;                    // kp == k when l > 0
      }
      const size_t off  = ((size_t)l * ZDIM + k) * PDIM + p;
      const float Wkl   = Wk - mw_in[off] * al_in[off];
      const float ERtZ  = rtz - Ezz * Wkl;
      const float s2    = 1.0f / (Ezz * tau);
      const float s0inv = 1.0f / tau0_in[l * ZDIM + k];
      const float s2p   = s2 + s0inv;
      const float z2    = ERtZ * ERtZ * tau / Ezz;       // Z_s^2
      const float logbf = 0.5f * (__logf(s2) - __logf(s2p) + z2 * (s0inv / s2p));
      const float logit = __logf(pi_in[(size_t)k * PDIM + p]) + logbf;
      logits[p] = logit;
      mLoc = logit;
      sLoc = 1.0f;
    }
    // wave32 online-softmax reduction
#pragma unroll
    for (int off = 16; off > 0; off >>= 1) {
      const float om = __shfl_xor(mLoc, off, 32);
      const float os = __shfl_xor(sLoc, off, 32);
      softmax_merge(mLoc, sLoc, om, os);
    }
    const int wv = threadIdx.x >> 5;
    const int ln = threadIdx.x & 31;
    if (ln == 0) { shm[wv] = mLoc; shs[wv] = sLoc; }
    __syncthreads();
    if (threadIdx.x == 0) {
      float bm = shm[0], bs = shs[0];
#pragma unroll
      for (int w = 1; w < 8; ++w) softmax_merge(bm, bs, shm[w], shs[w]);
      // lock-free merge of packed (m,S) into this step's global slot
      unsigned long long* slot = &msSlots[s];
      unsigned long long cur = atomicAdd(slot, 0ULL);
      for (;;) {
        float gm = __uint_as_float((unsigned)(cur & 0xffffffffu));
        float gs = __uint_as_float((unsigned)(cur >> 32));
        softmax_merge(gm, gs, bm, bs);
        const unsigned long long nv =
            ((unsigned long long)__float_as_uint(gs) << 32) | __float_as_uint(gm);
        const unsigned long long prev = atomicCAS(slot, cur, nv);
        if (prev == cur) break;
        cur = prev;
      }
    }
  }
}

// ---------------------------------------------------------------------------
extern "C" void kernel_launch(void* const* d_in, const int* in_sizes, int n_in,
                              void* d_out, int out_size, void* d_ws, size_t ws_size,
                              hipStream_t stream) {
  const float* data  = (const float*)d_in[0];   // [N,P]
  const float* mz    = (const float*)d_in[1];   // [N,Z]
  const float* varz  = (const float*)d_in[2];   // [Z,Z]
  const float* mw_in = (const float*)d_in[3];   // [L,Z,P]
  // d_in[4] = var_w: every entry is overwritten by the updates -> never read
  const float* al_in = (const float*)d_in[5];   // [L,Z,P]
  const float* tau0  = (const float*)d_in[6];   // [L,Z]
  const float* pi    = (const float*)d_in[7];   // [Z,P]
  const float* tau   = (const float*)d_in[8];   // scalar

  float* out    = (float*)d_out;                // concat(mw, vw, al)
  float* mw_out = out;
  float* vw_out = out + (size_t)LDIM * ZP;
  float* al_out = vw_out + LDIM * ZDIM;

  // workspace layout (floats): ~4.5 MB total
  float* ws     = (float*)d_ws;
  float* W      = ws;                            // [Z,P]   500000
  float* ZtD    = ws + ZP;                       // [Z,P]   500000
  float* RtZk   = ws + 2 * ZP;                   // [P]      50000
  float* logits = ws + 2 * ZP + PDIM;            // [P]      50000
  float* mzz    = ws + 2 * ZP + 2 * PDIM;        // [Z,Z]   (pad 128)
  float* mzPair = ws + 2 * ZP + 2 * PDIM + 128;  // [500][16][2] 16000
  unsigned long long* msSlots =
      (unsigned long long*)(ws + 2 * ZP + 2 * PDIM + 128 + 16000);  // 8B-aligned

  init_kernel<<<(ZP + 255) / 256, 256, 0, stream>>>(mz, mw_in, al_in, W,
                                                    mzPair, msSlots);

  // Phase A: TDM-staged WMMA GEMM (bandwidth bound: 200MB ~ 8.6us floor)
  ztd_wmma_tdm_kernel<<<(PDIM + COLS_PER_BLOCK - 1) / COLS_PER_BLOCK, 256, 0,
                        stream>>>(data, mzPair, ZtD);
  mzz_wmma_kernel<<<1, 32, 0, stream>>>(mzPair, varz, mzz);

  // Phase B: 50 dependent updates, one fused kernel each (+1 final finalize)
  const int blocks = (PDIM + 255) / 256;
  for (int s = 0; s <= NSTEP; ++s) {
    vb_step_kernel<<<blocks, 256, 0, stream>>>(s, mw_in, al_in, tau0, pi, tau,
                                               W, ZtD, RtZk, logits, mzz, msSlots,
                                               mw_out, vw_out, al_out);
  }
}